// MultiHeadHyperNet_15522011808517
// MI455X (gfx1250) — compile-verified
//
#include <hip/hip_runtime.h>
#include <math.h>

typedef float v2f __attribute__((ext_vector_type(2)));
typedef float v8f __attribute__((ext_vector_type(8)));

#define LN_EPS 1e-5f
#define NTILE_N 38          // 608 padded columns / 16
#define NCOLS   608         // 152 groups * 4 (3 real splits + 1 dummy per (c,t))
#define NVALID  600         // 150 groups * 4
#define NTASKS  (150*467)   // head2 outputs needed per (c,t): 384 splitW + 3 bias + 80 leaf

// ---------------------------------------------------------------------------
// Stage A: mean of X_train (deterministic two-pass)
// ---------------------------------------------------------------------------
__global__ __launch_bounds__(128) void mean_partial_kernel(const float* __restrict__ X,
                                                           float* __restrict__ part) {
    int d = threadIdx.x;            // 0..127
    int b = blockIdx.x;             // 0..199, 500 rows each
    size_t r0 = (size_t)b * 500;
    float s = 0.f;
    for (int r = 0; r < 500; ++r) s += X[(r0 + r) * 128 + d];
    part[b * 128 + d] = s;
}

__global__ __launch_bounds__(128) void mean_final_kernel(const float* __restrict__ part,
                                                         float* __restrict__ mean) {
    int d = threadIdx.x;
    float s = 0.f;
    for (int b = 0; b < 200; ++b) s += part[b * 128 + d];
    mean[d] = s * (1.f / 100000.f);
}

// ---------------------------------------------------------------------------
// Stage B1/B2: encoder layer:  out = relu(LN(W @ x + b))
// ---------------------------------------------------------------------------
__global__ __launch_bounds__(512) void enc_layer_kernel(const float* __restrict__ W,
                                                        const float* __restrict__ bias,
                                                        const float* __restrict__ g,
                                                        const float* __restrict__ beta,
                                                        const float* __restrict__ xin,
                                                        int inDim,
                                                        float* __restrict__ out) {
    __shared__ float sx[512];
    __shared__ float sy[512];
    __shared__ float sr[512];
    int j = threadIdx.x;
    if (j < inDim) sx[j] = xin[j];
    __syncthreads();
    const float* wr = W + (size_t)j * inDim;
    float acc = bias[j];
    for (int d = 0; d < inDim; ++d) acc += wr[d] * sx[d];
    sy[j] = acc;
    sr[j] = acc;
    __syncthreads();
    for (int off = 256; off > 0; off >>= 1) { if (j < off) sr[j] += sr[j + off]; __syncthreads(); }
    float m = sr[0] * (1.f / 512.f);
    __syncthreads();
    float cen = sy[j] - m;
    sr[j] = cen * cen;
    __syncthreads();
    for (int off = 256; off > 0; off >>= 1) { if (j < off) sr[j] += sr[j + off]; __syncthreads(); }
    float v = sr[0] * (1.f / 512.f);
    float hn = cen * rsqrtf(v + LN_EPS) * g[j] + beta[j];
    out[j] = fmaxf(hn, 0.f);
}

// ---------------------------------------------------------------------------
// Stage B3: per-class head layer 1:  hh[c] = relu(W1[c] @ h2 + b1[c])
// ---------------------------------------------------------------------------
__global__ __launch_bounds__(512) void head1_kernel(const float* __restrict__ W1,
                                                    const float* __restrict__ b1,
                                                    const float* __restrict__ h2,
                                                    float* __restrict__ hh) {
    __shared__ float sx[512];
    int j = threadIdx.x;
    int c = blockIdx.x;
    sx[j] = h2[j];
    __syncthreads();
    const float* wr = W1 + ((size_t)c * 512 + j) * 512;
    float acc = b1[c * 512 + j];
    for (int d = 0; d < 512; ++d) acc += wr[d] * sx[d];
    hh[c * 512 + j] = fmaxf(acc, 0.f);
}

// ---------------------------------------------------------------------------
// Stage B4: needed params = head_W2[c,p,:] . hh[c] + head_b2[c,p]
// One wave per output; scatter into SW / Bias / LeafLogit.
// Only splits i<3, their biases, and leaf logits are computed (rest unused).
// ---------------------------------------------------------------------------
__global__ __launch_bounds__(256) void head2_kernel(const float* __restrict__ W2,
                                                    const float* __restrict__ b2,
                                                    const float* __restrict__ hh,
                                                    float* __restrict__ SW,
                                                    float* __restrict__ BiasBuf,
                                                    float* __restrict__ LeafLogit) {
    int lane = threadIdx.x & 31;
    int w = blockIdx.x * 8 + (threadIdx.x >> 5);
    if (w >= NTASKS) return;
    int ct  = w / 467;
    int rin = w % 467;
    int c = ct / 15;
    int t = ct % 15;
    int r;
    if (rin < 384)      r = rin;                 // split_w rows i<3 : r = i*128 + d
    else if (rin < 387) r = 896 + (rin - 384);   // split_b i<3     : r = 7*128 + i
    else                r = 903 + (rin - 387);   // leaf logits     : r = 903 + l*10 + k
    int pidx = t * 983 + r;
    const float* wr = W2 + ((size_t)c * 14745 + pidx) * 512;
    const float* hv = hh + c * 512;
    float acc = 0.f;
    #pragma unroll 4
    for (int q = 0; q < 16; ++q) acc += wr[lane + 32 * q] * hv[lane + 32 * q];
    acc += __shfl_xor(acc, 16, 32);
    acc += __shfl_xor(acc, 8, 32);
    acc += __shfl_xor(acc, 4, 32);
    acc += __shfl_xor(acc, 2, 32);
    acc += __shfl_xor(acc, 1, 32);
    if (lane == 0) {
        float val = acc + b2[(size_t)c * 14745 + pidx];
        if (rin < 384)      SW[(size_t)(4 * ct + (rin >> 7)) * 128 + (rin & 127)] = val;
        else if (rin < 387) BiasBuf[4 * ct + (rin - 384)] = val;
        else                LeafLogit[ct * 80 + (rin - 387)] = val;
    }
}

// ---------------------------------------------------------------------------
// Stage C: scaled leaf matrix  SL[ct][l][k] = softmax(tw)[t] * softmax(leaf)[k] / 10
// ---------------------------------------------------------------------------
__global__ void slc_kernel(const float* __restrict__ tw,
                           const float* __restrict__ LeafLogit,
                           float* __restrict__ SL) {
    int tid = blockIdx.x * blockDim.x + threadIdx.x;
    if (tid >= 152 * 8) return;
    int ct = tid >> 3, l = tid & 7;
    float* dst = SL + ct * 80 + l * 10;
    if (ct >= 150) { for (int k = 0; k < 10; ++k) dst[k] = 0.f; return; }
    int t = ct % 15;
    float mx = tw[0];
    for (int i = 1; i < 15; ++i) mx = fmaxf(mx, tw[i]);
    float den = 0.f, wt = 0.f;
    for (int i = 0; i < 15; ++i) { float e = __expf(tw[i] - mx); den += e; if (i == t) wt = e; }
    wt /= den;
    const float* lg = LeafLogit + ct * 80 + l * 10;
    float m2 = lg[0];
    for (int k = 1; k < 10; ++k) m2 = fmaxf(m2, lg[k]);
    float e[10], s = 0.f;
    for (int k = 0; k < 10; ++k) { e[k] = __expf(lg[k] - m2); s += e[k]; }
    float sc = wt * 0.1f / s;
    for (int k = 0; k < 10; ++k) dst[k] = e[k] * sc;
}

// ---------------------------------------------------------------------------
// Stage prep: swizzle split weights into per-lane WMMA B-operand layout.
// Bbuf[nt][ks][lane] = float2{W[k][n], W[k+1][n]}, k = 4*ks + (lane<16?0:2),
// n = nt*16 + lane%16, zero for dummy columns.
// ---------------------------------------------------------------------------
__global__ void bprep_kernel(const float* __restrict__ SW, float* __restrict__ Bbuf) {
    int id = blockIdx.x * blockDim.x + threadIdx.x;
    if (id >= NTILE_N * 32 * 32) return;
    int lane = id & 31;
    int ks = (id >> 5) & 31;
    int nt = id >> 10;
    int n  = nt * 16 + (lane & 15);
    int k0 = ks * 4 + ((lane >> 4) << 1);
    float2 v = make_float2(0.f, 0.f);
    if (n < NVALID && (n & 3) != 3) {
        v.x = SW[(size_t)n * 128 + k0];
        v.y = SW[(size_t)n * 128 + k0 + 1];
    }
    ((float2*)Bbuf)[id] = v;
}

__global__ void zero_bias_kernel(float* __restrict__ BiasBuf) {
    int t = blockIdx.x * blockDim.x + threadIdx.x;
    if (t < 640) BiasBuf[t] = 0.f;
}

// ---------------------------------------------------------------------------
// Main fused kernel: per wave, 16 test rows.
//   X tile staged into LDS with GLOBAL_LOAD_ASYNC_TO_LDS_B128 (ASYNCcnt),
//   GEMM 16x128 * 128x608 via V_WMMA_F32_16X16X4_F32 (32 K-steps per N-tile),
//   then sigmoid + soft-tree leaf products + scaled-leaf accumulation,
//   final cross-lane reduction and output store.
// ---------------------------------------------------------------------------
__global__ __launch_bounds__(128) void forest_kernel(const float* __restrict__ Xt,
                                                     const float* __restrict__ Bbuf,
                                                     const float* __restrict__ BiasBuf,
                                                     const float* __restrict__ SL,
                                                     float* __restrict__ out,
                                                     int nTiles) {
    __shared__ float Atile[4 * 16 * 132];   // 4 waves, 16 rows, pitch 132 (bank-conflict-free)
    int lane = threadIdx.x & 31;
    int wid  = threadIdx.x >> 5;
    int tile = blockIdx.x * 4 + wid;
    bool active = tile < nTiles;
    float* Aw = Atile + wid * (16 * 132);

    if (active) {
        int base = tile * 16;
        // Async memory->LDS DMA: one X row (512B across the wave) per op.
        #pragma unroll
        for (int r = 0; r < 16; ++r) {
            const float* gaddr = Xt + (size_t)(base + r) * 128 + lane * 4;
            unsigned laddr = (unsigned)(size_t)(Aw + r * 132 + lane * 4);
            asm volatile("global_load_async_to_lds_b128 %0, %1, off"
                         :: "v"(laddr), "v"(gaddr) : "memory");
        }
        asm volatile("s_wait_asynccnt 0x0" ::: "memory");
    }
    __syncthreads();
    if (!active) return;

    int nHalf = lane >> 4;          // 0: K-pair {0,1}, 1: K-pair {2,3}
    int nIdx  = lane & 15;          // M for A-operand, N for B/C/D
    int p     = lane & 3;           // position within (c,t) group: handles leaves p, p+4
    int gInTile  = nIdx >> 2;       // 4 (c,t) groups per 16-wide tile
    int baseLane = lane & ~3;

    float priv[8][10];
    #pragma unroll
    for (int s = 0; s < 8; ++s)
        #pragma unroll
        for (int k = 0; k < 10; ++k) priv[s][k] = 0.f;

    const v2f* Bv = (const v2f*)Bbuf;

    for (int nt = 0; nt < NTILE_N; ++nt) {
        v8f acc = {0.f, 0.f, 0.f, 0.f, 0.f, 0.f, 0.f, 0.f};
        const v2f* bp = Bv + (size_t)nt * 1024 + lane;
        if (nt + 1 < NTILE_N)
            __builtin_prefetch((const void*)(bp + 1024), 0, 3);   // next B tile
        #pragma unroll 8
        for (int ks = 0; ks < 32; ++ks) {
            float2 av = *(const float2*)(Aw + nIdx * 132 + ks * 4 + nHalf * 2);
            v2f a; a.x = av.x; a.y = av.y;
            v2f b = bp[ks * 32];
            acc = __builtin_amdgcn_wmma_f32_16x16x4_f32(false, a, false, b,
                                                        (short)0, acc, false, false);
        }
        int n = nt * 16 + nIdx;
        float biasv = BiasBuf[n];
        int ct = nt * 4 + gInTile;
        float slA[10], slB[10];
        const float* slp = SL + ct * 80 + p * 10;
        #pragma unroll
        for (int k = 0; k < 10; ++k) { slA[k] = slp[k]; slB[k] = slp[40 + k]; }

        #pragma unroll
        for (int s = 0; s < 8; ++s) {
            float y   = acc[s] + biasv;
            float sig = 1.f / (1.f + __expf(-y));
            float r0 = __shfl(sig, baseLane + 0, 32);
            float r1 = __shfl(sig, baseLane + 1, 32);
            float r2 = __shfl(sig, baseLane + 2, 32);
            float f01 = ((p & 1) ? (1.f - r0) : r0) * ((p & 2) ? (1.f - r1) : r1);
            float lpA = f01 * r2;           // leaf p   (bit2 = 0)
            float lpB = f01 * (1.f - r2);   // leaf p+4 (bit2 = 1)
            #pragma unroll
            for (int k = 0; k < 10; ++k)
                priv[s][k] += lpA * slA[k] + lpB * slB[k];
        }
    }

    int rowBase = tile * 16;
    #pragma unroll
    for (int s = 0; s < 8; ++s) {
        #pragma unroll
        for (int k = 0; k < 10; ++k) {
            float v = priv[s][k];
            v += __shfl_xor(v, 1, 32);
            v += __shfl_xor(v, 2, 32);
            v += __shfl_xor(v, 4, 32);
            v += __shfl_xor(v, 8, 32);
            if (lane == 0)  out[(size_t)(rowBase + s)     * 10 + k] = v;
            if (lane == 16) out[(size_t)(rowBase + s + 8) * 10 + k] = v;
        }
    }
}

// ---------------------------------------------------------------------------
extern "C" void kernel_launch(void* const* d_in, const int* in_sizes, int n_in,
                              void* d_out, int out_size, void* d_ws, size_t ws_size,
                              hipStream_t stream) {
    const float* X_train = (const float*)d_in[0];
    const float* X_test  = (const float*)d_in[1];
    const float* enc_W1  = (const float*)d_in[2];
    const float* enc_b1  = (const float*)d_in[3];
    const float* ln1_g   = (const float*)d_in[4];
    const float* ln1_b   = (const float*)d_in[5];
    const float* enc_W2  = (const float*)d_in[6];
    const float* enc_b2  = (const float*)d_in[7];
    const float* ln2_g   = (const float*)d_in[8];
    const float* ln2_b   = (const float*)d_in[9];
    const float* head_W1 = (const float*)d_in[10];
    const float* head_b1 = (const float*)d_in[11];
    const float* head_W2 = (const float*)d_in[12];
    const float* head_b2 = (const float*)d_in[13];
    const float* tree_w  = (const float*)d_in[14];
    float* out = (float*)d_out;

    float* wsf       = (float*)d_ws;
    float* part      = wsf;                     // 200*128 = 25600
    float* mean      = part + 25600;            // 128
    float* h1        = mean + 128;              // 512
    float* h2        = h1 + 512;                // 512
    float* hh        = h2 + 512;                // 10*512 = 5120
    float* SW        = hh + 5120;               // 608*128 = 77824
    float* BiasBuf   = SW + 77824;              // 640 (608 used, padded)
    float* LeafLogit = BiasBuf + 640;           // 152*80 = 12160
    float* SL        = LeafLogit + 12160;       // 152*80 = 12160
    float* Bbuf      = SL + 12160;              // 38*32*32*2 = 77824 (float2-aligned)

    zero_bias_kernel<<<1, 640, 0, stream>>>(BiasBuf);
    mean_partial_kernel<<<200, 128, 0, stream>>>(X_train, part);
    mean_final_kernel<<<1, 128, 0, stream>>>(part, mean);
    enc_layer_kernel<<<1, 512, 0, stream>>>(enc_W1, enc_b1, ln1_g, ln1_b, mean, 128, h1);
    enc_layer_kernel<<<1, 512, 0, stream>>>(enc_W2, enc_b2, ln2_g, ln2_b, h1, 512, h2);
    head1_kernel<<<10, 512, 0, stream>>>(head_W1, head_b1, h2, hh);
    head2_kernel<<<(NTASKS + 7) / 8, 256, 0, stream>>>(head_W2, head_b2, hh,
                                                       SW, BiasBuf, LeafLogit);
    slc_kernel<<<5, 256, 0, stream>>>(tree_w, LeafLogit, SL);
    bprep_kernel<<<(NTILE_N * 1024 + 255) / 256, 256, 0, stream>>>(SW, Bbuf);

    int nTiles = 100000 / 16;                   // 6250, exact
    forest_kernel<<<(nTiles + 3) / 4, 128, 0, stream>>>(X_test, Bbuf, BiasBuf, SL,
                                                        out, nTiles);
}